// deephi_Correlation1D_Elemwise_43963285242020
// MI455X (gfx1250) — compile-verified
//
#include <hip/hip_runtime.h>

// Problem constants (fixed by the reference)
#define CC   32
#define HH   128
#define WW   512
#define PADS 40
#define DD   (PADS + 1)

// Native ext-vector types (HIP_vector_type structs are rejected by the builtins)
typedef int   v4i __attribute__((ext_vector_type(4)));
typedef float v4f __attribute__((ext_vector_type(4)));

#if defined(__AMDGCN__) && __has_builtin(__builtin_amdgcn_global_load_async_to_lds_b128)
#define USE_ASYNC_LDS 1
// Builtin prototype: (int4 addrspace(1)*, int4 addrspace(3)*, imm offset, imm cpol)
typedef __attribute__((address_space(1))) v4i* gbl_v4i_p;
typedef __attribute__((address_space(3))) v4i* lds_v4i_p;
#else
#define USE_ASYNC_LDS 0
#endif

// out[c,d,h,w] = in1[c,h,w] * in2[c,h,w-d]  (0 when w < d)
// One block per (c,h) row pair. 128 threads, thread t owns float4 at w = 4t.
__global__ __launch_bounds__(128) void corr1d_cost_volume(
    const float* __restrict__ in1,
    const float* __restrict__ in2,
    float* __restrict__ out) {
  __shared__ __align__(16) float lds1[WW];          // in1 row
  __shared__ __align__(16) float lds2[PADS + WW];   // zero pad + in2 row

  const int tid = threadIdx.x;            // 0..127
  const int row = blockIdx.x;             // c*H + h
  const int c   = row >> 7;               // /128
  const int h   = row & (HH - 1);

  const float* __restrict__ row1 = in1 + (size_t)row * WW;
  const float* __restrict__ row2 = in2 + (size_t)row * WW;

  // Zero the left pad (threads 0..39)
  if (tid < PADS) lds2[tid] = 0.0f;

  const int w0 = tid * 4;

#if USE_ASYNC_LDS
  // CDNA5 async global->LDS copy (tracked by ASYNCcnt). Each lane moves 16B.
  __builtin_amdgcn_global_load_async_to_lds_b128(
      (gbl_v4i_p)(row1 + w0), (lds_v4i_p)&lds1[w0],
      /*offset=*/0, /*cpol=*/0);
  __builtin_amdgcn_global_load_async_to_lds_b128(
      (gbl_v4i_p)(row2 + w0), (lds_v4i_p)&lds2[PADS + w0],
      /*offset=*/0, /*cpol=*/0);
#if __has_builtin(__builtin_amdgcn_s_wait_asynccnt)
  __builtin_amdgcn_s_wait_asynccnt(0);
#else
  asm volatile("s_wait_asynccnt 0" ::: "memory");
#endif
#else
  // Fallback staging path (no async builtin available on this toolchain)
  *reinterpret_cast<v4f*>(&lds1[w0]) =
      *reinterpret_cast<const v4f*>(row1 + w0);
  *reinterpret_cast<v4f*>(&lds2[PADS + w0]) =
      *reinterpret_cast<const v4f*>(row2 + w0);
#endif

  __syncthreads();

  // in1 float4 lives in registers across all 41 disparity planes (41x reuse)
  const v4f a = *reinterpret_cast<const v4f*>(&lds1[w0]);

  // Base of the d=0 plane for this (c,h); planes are H*W apart
  float* __restrict__ orow = out + (((size_t)c * DD) * HH + h) * (size_t)WW;
  const size_t dstride = (size_t)HH * WW;

#pragma unroll 4
  for (int d = 0; d <= PADS; ++d) {
    // Shifted in2 read: pad guarantees in-range, branch-free (LDS is cheap)
    const float* __restrict__ b = &lds2[PADS + w0 - d];
    v4f r;
    r.x = a.x * b[0];
    r.y = a.y * b[1];
    r.z = a.z * b[2];
    r.w = a.w * b[3];
    // 344 MB streamed output: non-temporal 128-bit store (TH=NT), keep L2 clean
    __builtin_nontemporal_store(r, reinterpret_cast<v4f*>(orow + w0));
    orow += dstride;
  }
}

extern "C" void kernel_launch(void* const* d_in, const int* in_sizes, int n_in,
                              void* d_out, int out_size, void* d_ws, size_t ws_size,
                              hipStream_t stream) {
  (void)in_sizes; (void)n_in; (void)out_size; (void)d_ws; (void)ws_size;
  const float* in1 = (const float*)d_in[0];
  const float* in2 = (const float*)d_in[1];
  float* out = (float*)d_out;
  // d_in[2] is pad_size (== 40, fixed by the reference); compile-time constant here.
  dim3 grid(CC * HH);   // 4096 blocks, one per (c,h) row
  dim3 block(128);      // 4 wave32 waves; thread t owns float4 at w=4t
  hipLaunchKernelGGL(corr1d_cost_volume, grid, block, 0, stream, in1, in2, out);
}